// TrafficGNN_74191265071850
// MI455X (gfx1250) — compile-verified
//
#include <hip/hip_runtime.h>
#include <hip/hip_bf16.h>

// ---------------------------------------------------------------------------
// GCN forward on gfx1250 (MI455X):
//   h1 = MP(x@W1) ; BN+ReLU ; h2 = MP(a@W2) ; log_softmax
// GEMMs use V_WMMA_F32_16X16X4_F32 (exact fp32 matrix path on CDNA5).
// ---------------------------------------------------------------------------

typedef __attribute__((ext_vector_type(2))) float v2f;
typedef __attribute__((ext_vector_type(8))) float v8f;

#define IN_F   128
#define BN_EPS 1e-5f

// ------------------------------ utility ------------------------------------

__global__ void zero_f32_kernel(float* __restrict__ p, long n) {
    long tid = (long)blockIdx.x * blockDim.x + threadIdx.x;
    if (tid < n) p[tid] = 0.0f;
}

__global__ void deg_scatter_kernel(const int* __restrict__ col,
                                   const float* __restrict__ ew,
                                   float* __restrict__ deg, int E) {
    int e = blockIdx.x * blockDim.x + threadIdx.x;
    if (e >= E) return;
    atomicAdd(&deg[col[e]], ew[e]);
}

// dinv computed in place over deg; self-loop weight (+1) folded in here.
__global__ void dinv_kernel(float* __restrict__ deg, int n) {
    int i = blockIdx.x * blockDim.x + threadIdx.x;
    if (i >= n) return;
    deg[i] = rsqrtf(deg[i] + 1.0f);
}

// ------------------------------ WMMA GEMM ----------------------------------
// H[nrows x NCOL] = X[nrows x 128] @ W[128 x NCOL]
// block = 256 threads (8 waves); each wave computes one 16-row M tile.
// W staged in LDS; A fragments streamed from global per K-block of 4.

template <int NCOL>
__global__ void gemm_wmma_f32_kernel(const float* __restrict__ X,
                                     const float* __restrict__ W,
                                     float* __restrict__ H, int nrows) {
    __shared__ float Wlds[IN_F * NCOL];
    for (int i = threadIdx.x; i < IN_F * NCOL; i += blockDim.x) Wlds[i] = W[i];
    __syncthreads();

    const int wave = threadIdx.x >> 5;
    const int lane = threadIdx.x & 31;
    const int tile = blockIdx.x * 8 + wave;
    if (tile * 16 + 16 > nrows) return;   // full tiles only; tail kernel covers rest

    const int m0 = tile * 16;
    const int lr = lane & 15;   // row (A) / col (B,C,D) within tile
    const int hi = lane >> 4;   // half-wave selector

    constexpr int NT = NCOL / 16;
    v8f acc[NT];
    {
        v8f z = {};
#pragma unroll
        for (int n = 0; n < NT; ++n) acc[n] = z;
    }

    // A row base for this lane (float2 = the 2 K-values this lane holds per block)
    const float2* __restrict__ arow =
        (const float2*)(X + (size_t)(m0 + lr) * IN_F);

#pragma unroll 4
    for (int kb = 0; kb < IN_F / 4; ++kb) {
        const int k0 = kb * 4;
        // A 16x4 f32 layout: lane<16 -> K=k0,k0+1 ; lane>=16 -> K=k0+2,k0+3
        float2 a2 = arow[kb * 2 + hi];
        v2f a; a.x = a2.x; a.y = a2.y;

        const int rowA = (k0 + 2 * hi) * NCOL;      // B row for vgpr0
        const int rowB = rowA + NCOL;               // B row for vgpr1
#pragma unroll
        for (int n = 0; n < NT; ++n) {
            const int n0 = n * 16;
            v2f b;
            b.x = Wlds[rowA + n0 + lr];
            b.y = Wlds[rowB + n0 + lr];
            acc[n] = __builtin_amdgcn_wmma_f32_16x16x4_f32(
                false, a, false, b, (short)0, acc[n], false, false);
        }
    }

    // C/D layout: vgpr v, lane l -> D[M = v + 8*(l>>4)][N = l&15]
    float* __restrict__ hp = H + (size_t)(m0 + 8 * hi) * NCOL + lr;
#pragma unroll
    for (int n = 0; n < NT; ++n) {
        const int n0 = n * 16;
#pragma unroll
        for (int v = 0; v < 8; ++v) {
            hp[(size_t)v * NCOL + n0] = acc[n][v];
        }
    }
}

// scalar tail for nrows % 16 (not hit when N is a multiple of 16)
template <int NCOL>
__global__ void gemm_tail_kernel(const float* __restrict__ X,
                                 const float* __restrict__ W,
                                 float* __restrict__ H, int startRow, int nrows) {
    int tid = blockIdx.x * blockDim.x + threadIdx.x;
    int r = startRow + tid / NCOL;
    int c = tid % NCOL;
    if (r >= nrows) return;
    float s = 0.0f;
    for (int k = 0; k < IN_F; ++k)
        s += X[(size_t)r * IN_F + k] * W[(size_t)k * NCOL + c];
    H[(size_t)r * NCOL + c] = s;
}

// ------------------------- message passing ---------------------------------
// out[i] = dinv[i]^2 * H[i] + bias   (self-loop term + bias as initializer)

template <int F>
__global__ void mp_init_kernel(const float* __restrict__ H,
                               const float* __restrict__ dinv,
                               const float* __restrict__ bias,
                               float* __restrict__ out, int n) {
    constexpr int F4 = F / 4;
    long tid = (long)blockIdx.x * blockDim.x + threadIdx.x;
    if (tid >= (long)n * F4) return;
    int i  = (int)(tid / F4);
    int c4 = (int)(tid % F4);
    float d = dinv[i];
    float dd = d * d;
    float4 h = ((const float4*)H)[tid];
    float4 b = ((const float4*)bias)[c4];
    float4 o;
    o.x = fmaf(h.x, dd, b.x);
    o.y = fmaf(h.y, dd, b.y);
    o.z = fmaf(h.z, dd, b.z);
    o.w = fmaf(h.w, dd, b.w);
    ((float4*)out)[tid] = o;
}

// out[col] += dinv[row]*ew*dinv[col] * H[row] ; one thread per (edge, float4)
template <int F>
__global__ void mp_edge_kernel(const int* __restrict__ row,
                               const int* __restrict__ col,
                               const float* __restrict__ ew,
                               const float* __restrict__ dinv,
                               const float* __restrict__ H,
                               float* __restrict__ out, int E) {
    constexpr int F4 = F / 4;
    long tid = (long)blockIdx.x * blockDim.x + threadIdx.x;
    if (tid >= (long)E * F4) return;
    int e  = (int)(tid / F4);
    int c4 = (int)(tid % F4);
    int r = row[e];
    int c = col[e];
    float nrm = dinv[r] * ew[e] * dinv[c];
    float4 h = ((const float4*)H)[(size_t)r * F4 + c4];
    float* o = out + (size_t)c * F + c4 * 4;
    atomicAdd(o + 0, h.x * nrm);
    atomicAdd(o + 1, h.y * nrm);
    atomicAdd(o + 2, h.z * nrm);
    atomicAdd(o + 3, h.w * nrm);
}

// ------------------------------ batch norm ---------------------------------

__global__ void bn_stats_kernel(const float* __restrict__ h,
                                float* __restrict__ stats, int n) {
    int c = threadIdx.x;   // 128 threads, one per channel
    float s = 0.0f, s2 = 0.0f;
    for (int r = blockIdx.x; r < n; r += gridDim.x) {
        float v = h[(size_t)r * 128 + c];
        s += v;
        s2 += v * v;
    }
    atomicAdd(&stats[c], s);
    atomicAdd(&stats[128 + c], s2);
}

__global__ void bn_finalize_kernel(float* __restrict__ stats,
                                   const float* __restrict__ gamma,
                                   const float* __restrict__ beta, int n) {
    int c = threadIdx.x;
    float inv_n = 1.0f / (float)n;
    float mu  = stats[c] * inv_n;
    float var = stats[128 + c] * inv_n - mu * mu;
    float inv = rsqrtf(var + BN_EPS);
    float sc = inv * gamma[c];
    stats[256 + c] = sc;               // scale
    stats[384 + c] = beta[c] - mu * sc; // shift
}

__global__ void bn_apply_relu_kernel(float* __restrict__ h,
                                     const float* __restrict__ stats,
                                     long total4) {
    long tid = (long)blockIdx.x * blockDim.x + threadIdx.x;
    if (tid >= total4) return;
    int c = (int)(tid & 31) * 4;       // 32 float4 per 128-wide row
    float4 v = ((const float4*)h)[tid];
    float s0 = stats[256 + c], s1 = stats[257 + c],
          s2 = stats[258 + c], s3 = stats[259 + c];
    float t0 = stats[384 + c], t1 = stats[385 + c],
          t2 = stats[386 + c], t3 = stats[387 + c];
    v.x = fmaxf(0.0f, fmaf(v.x, s0, t0));
    v.y = fmaxf(0.0f, fmaf(v.y, s1, t1));
    v.z = fmaxf(0.0f, fmaf(v.z, s2, t2));
    v.w = fmaxf(0.0f, fmaf(v.w, s3, t3));
    ((float4*)h)[tid] = v;
}

// ------------------------------ log-softmax --------------------------------
// one wave32 per 64-wide row, in place. Lane l handles cols l and l+32.

__global__ void logsoftmax64_kernel(float* __restrict__ out, int n) {
    int wave = threadIdx.x >> 5;
    int lane = threadIdx.x & 31;
    int r = blockIdx.x * (blockDim.x >> 5) + wave;
    if (r >= n) return;
    float* p = out + (size_t)r * 64;
    float a = p[lane];
    float b = p[lane + 32];
    float m = fmaxf(a, b);
#pragma unroll
    for (int off = 16; off > 0; off >>= 1)
        m = fmaxf(m, __shfl_xor(m, off, 32));
    float s = expf(a - m) + expf(b - m);
#pragma unroll
    for (int off = 16; off > 0; off >>= 1)
        s += __shfl_xor(s, off, 32);
    float lse = m + logf(s);
    p[lane]      = a - lse;
    p[lane + 32] = b - lse;
}

// ------------------------------ launcher -----------------------------------

extern "C" void kernel_launch(void* const* d_in, const int* in_sizes, int n_in,
                              void* d_out, int out_size, void* d_ws, size_t ws_size,
                              hipStream_t stream) {
    const int N = in_sizes[0] / IN_F;   // 100000
    const int E = in_sizes[2];          // 1600000

    const float* x     = (const float*)d_in[0];
    const int*   ei    = (const int*)d_in[1];
    const float* ew    = (const float*)d_in[2];
    const float* W1    = (const float*)d_in[3];
    const float* b1    = (const float*)d_in[4];
    const float* gamma = (const float*)d_in[5];
    const float* beta  = (const float*)d_in[6];
    const float* W2    = (const float*)d_in[7];
    const float* b2    = (const float*)d_in[8];
    float* out = (float*)d_out;

    const int* row = ei;
    const int* col = ei + E;

    // workspace layout
    float* H1    = (float*)d_ws;                 // N*128
    float* O1    = H1 + (size_t)N * 128;         // N*128 (MP1 out -> activations)
    float* H2    = O1 + (size_t)N * 128;         // N*64
    float* deg   = H2 + (size_t)N * 64;          // N  (deg, then dinv in place)
    float* stats = deg + N;                      // 512: sum, sumsq, scale, shift

    const int T = 256;

    // 1) zero deg + stats (contiguous)
    {
        long n = (long)N + 512;
        zero_f32_kernel<<<(int)((n + T - 1) / T), T, 0, stream>>>(deg, n);
    }
    // 2) degree scatter, 3) dinv (self loop folded in)
    deg_scatter_kernel<<<(E + T - 1) / T, T, 0, stream>>>(col, ew, deg, E);
    dinv_kernel<<<(N + T - 1) / T, T, 0, stream>>>(deg, N);

    // 4) H1 = x @ W1 (WMMA fp32)
    {
        int tiles = N / 16;
        if (tiles > 0)
            gemm_wmma_f32_kernel<128><<<(tiles + 7) / 8, T, 0, stream>>>(x, W1, H1, N);
        int rem = N - tiles * 16;
        if (rem > 0)
            gemm_tail_kernel<128><<<(rem * 128 + T - 1) / T, T, 0, stream>>>(
                x, W1, H1, tiles * 16, N);
    }

    // 5-6) message passing layer 1 into O1
    {
        long t = (long)N * 32;
        mp_init_kernel<128><<<(int)((t + T - 1) / T), T, 0, stream>>>(H1, deg, b1, O1, N);
        long te = (long)E * 32;
        mp_edge_kernel<128><<<(int)((te + T - 1) / T), T, 0, stream>>>(
            row, col, ew, deg, H1, O1, E);
    }

    // 7-9) batch norm (training-mode batch stats) + ReLU, in place on O1
    bn_stats_kernel<<<256, 128, 0, stream>>>(O1, stats, N);
    bn_finalize_kernel<<<1, 128, 0, stream>>>(stats, gamma, beta, N);
    {
        long t = (long)N * 32;
        bn_apply_relu_kernel<<<(int)((t + T - 1) / T), T, 0, stream>>>(O1, stats, t);
    }

    // 10) H2 = a @ W2 (WMMA fp32)
    {
        int tiles = N / 16;
        if (tiles > 0)
            gemm_wmma_f32_kernel<64><<<(tiles + 7) / 8, T, 0, stream>>>(O1, W2, H2, N);
        int rem = N - tiles * 16;
        if (rem > 0)
            gemm_tail_kernel<64><<<(rem * 64 + T - 1) / T, T, 0, stream>>>(
                O1, W2, H2, tiles * 16, N);
    }

    // 11-12) message passing layer 2 directly into d_out
    {
        long t = (long)N * 16;
        mp_init_kernel<64><<<(int)((t + T - 1) / T), T, 0, stream>>>(H2, deg, b2, out, N);
        long te = (long)E * 16;
        mp_edge_kernel<64><<<(int)((te + T - 1) / T), T, 0, stream>>>(
            row, col, ew, deg, H2, out, E);
    }

    // 13) log-softmax over 64 classes, in place
    logsoftmax64_kernel<<<(N + 7) / 8, T, 0, stream>>>(out, N);
}